// Net_graph_11390253269721
// MI455X (gfx1250) — compile-verified
//
#include <hip/hip_runtime.h>
#include <hip/hip_bf16.h>

typedef __attribute__((ext_vector_type(2))) float v2f;
typedef __attribute__((ext_vector_type(8))) float v8f;

#define WAVES_PER_BLOCK 8  // 256 threads = 8 wave32

// ---------------------------------------------------------------------------
// Degree / normalization
// ---------------------------------------------------------------------------
__global__ void k_deg_init(float* deg, int n) {
    int i = blockIdx.x * blockDim.x + threadIdx.x;
    if (i < n) deg[i] = 1.0f;  // +1 self loop
}

__global__ void k_deg_accum(const int* __restrict__ dst, float* __restrict__ deg, int n_edges) {
    int e = blockIdx.x * blockDim.x + threadIdx.x;
    if (e < n_edges) atomicAdd(&deg[dst[e]], 1.0f);
}

__global__ void k_dinv(float* deg, int n) {
    int i = blockIdx.x * blockDim.x + threadIdx.x;
    if (i < n) deg[i] = rsqrtf(deg[i]);  // deg >= 1 always
}

// ---------------------------------------------------------------------------
// Dense transforms via V_WMMA_F32_16X16X4_F32 (full f32, matches reference).
// One wave = one 16-row node tile. All operand feeds are branch-free
// (unconditional loads + v_cndmask selects); the only divergence control is
// the wave-uniform full/partial-tile split, so EXEC is all-ones at the WMMA.
// ---------------------------------------------------------------------------

// Layer 1: xw[N,16] = x[N,2] @ W1[2,16]; K padded 2 -> 4 with zeros.
__global__ void k_gemm_in2(const float* __restrict__ x, const float* __restrict__ W,
                           float* __restrict__ xw, int n_nodes, int n_tiles) {
    const int lane = threadIdx.x & 31;
    const int wave = threadIdx.x >> 5;
    const int tile = blockIdx.x * WAVES_PER_BLOCK + wave;
    if (tile >= n_tiles) return;              // wave-uniform exit

    const int base = tile * 16;
    const int half = lane >> 4;               // 0: lanes 0-15, 1: lanes 16-31
    const int n    = lane & 15;
    const int row  = base + n;                // A-matrix M index
    const bool full = (base + 16 <= n_nodes); // wave-uniform

    // B 4x16 f32: K rows 0,1 live in half 0; rows 2,3 are zero padding.
    // Load unconditionally (address valid for every lane), select via cndmask.
    const float w0 = W[n];
    const float w1 = W[16 + n];
    v2f b;
    b.x = half ? 0.0f : w0;
    b.y = half ? 0.0f : w1;

    // A 16x4 f32: half 0 holds K={0,1} = x[row, 0:2]; half 1 holds zero pad.
    const int rowc = full ? row : (row < n_nodes ? row : n_nodes - 1);
    const v2f xv = *(const v2f*)(x + (size_t)rowc * 2);  // b64 load
    v2f a;
    a.x = half ? 0.0f : xv.x;
    a.y = half ? 0.0f : xv.y;

    v8f c = {};
    c = __builtin_amdgcn_wmma_f32_16x16x4_f32(false, a, false, b, (short)0, c,
                                              false, false);

    if (full) {
        #pragma unroll
        for (int v = 0; v < 8; ++v)           // M = v + 8*half, N = lane&15
            xw[(size_t)(base + v + 8 * half) * 16 + n] = c[v];
    } else {
        #pragma unroll
        for (int v = 0; v < 8; ++v) {
            int r = base + v + 8 * half;
            if (r < n_nodes) xw[(size_t)r * 16 + n] = c[v];
        }
    }
}

// 16->16: xw[N,16] = act(h)[N,16] @ W[16,16], chained over 4 K-slices of 4.
template <bool RELU_IN>
__global__ void k_gemm16(const float* __restrict__ h, const float* __restrict__ W,
                         float* __restrict__ xw, int n_nodes, int n_tiles) {
    const int lane = threadIdx.x & 31;
    const int wave = threadIdx.x >> 5;
    const int tile = blockIdx.x * WAVES_PER_BLOCK + wave;
    if (tile >= n_tiles) return;              // wave-uniform exit

    const int base = tile * 16;
    const int half = lane >> 4;
    const int n    = lane & 15;
    const int row  = base + n;
    const bool full = (base + 16 <= n_nodes); // wave-uniform
    const int rowc = full ? row : (row < n_nodes ? row : n_nodes - 1);

    // Preload four B k-slices (W is 16x16, L2-resident), unconditional loads.
    v2f b[4];
    #pragma unroll
    for (int s = 0; s < 4; ++s) {
        const int k0 = 4 * s + 2 * half;
        b[s].x = W[(k0 + 0) * 16 + n];
        b[s].y = W[(k0 + 1) * 16 + n];
    }

    // Preload four A k-slices as b64 loads (row*16 + 4s + 2*half is even).
    const v2f* hrow = (const v2f*)(h + (size_t)rowc * 16);
    v2f a[4];
    #pragma unroll
    for (int s = 0; s < 4; ++s) {
        v2f av = hrow[2 * s + half];
        if (RELU_IN) { av.x = fmaxf(av.x, 0.0f); av.y = fmaxf(av.y, 0.0f); }
        a[s] = av;
    }

    v8f c = {};
    #pragma unroll
    for (int s = 0; s < 4; ++s)
        c = __builtin_amdgcn_wmma_f32_16x16x4_f32(false, a[s], false, b[s],
                                                  (short)0, c, false, false);

    if (full) {
        #pragma unroll
        for (int v = 0; v < 8; ++v)
            xw[(size_t)(base + v + 8 * half) * 16 + n] = c[v];
    } else {
        #pragma unroll
        for (int v = 0; v < 8; ++v) {
            int r = base + v + 8 * half;
            if (r < n_nodes) xw[(size_t)r * 16 + n] = c[v];
        }
    }
}

// 16->1: xw3[i] = relu(h[i,:]) . W3
__global__ void k_gemm_out1(const float* __restrict__ h, const float* __restrict__ W3,
                            float* __restrict__ xw3, int n_nodes) {
    int i = blockIdx.x * blockDim.x + threadIdx.x;
    if (i >= n_nodes) return;
    const float4* hr = (const float4*)(h + (size_t)i * 16);
    float acc = 0.0f;
    #pragma unroll
    for (int q = 0; q < 4; ++q) {
        float4 v = hr[q];
        acc += fmaxf(v.x, 0.0f) * W3[q * 4 + 0];
        acc += fmaxf(v.y, 0.0f) * W3[q * 4 + 1];
        acc += fmaxf(v.z, 0.0f) * W3[q * 4 + 2];
        acc += fmaxf(v.w, 0.0f) * W3[q * 4 + 3];
    }
    xw3[i] = acc;
}

// ---------------------------------------------------------------------------
// Self-loop init + edge scatter
// ---------------------------------------------------------------------------
__global__ void k_init16(const float* __restrict__ xw, const float* __restrict__ dinv,
                         const float* __restrict__ bias, float* __restrict__ out, int n_nodes) {
    int t = blockIdx.x * blockDim.x + threadIdx.x;
    int i = t >> 4, f = t & 15;
    if (i < n_nodes) {
        float di = dinv[i];
        out[t] = di * di * xw[t] + bias[f];
    }
}

__global__ void k_init1(const float* __restrict__ xw3, const float* __restrict__ dinv,
                        const float* __restrict__ bias, float* __restrict__ out, int n_nodes) {
    int i = blockIdx.x * blockDim.x + threadIdx.x;
    if (i < n_nodes) {
        float di = dinv[i];
        out[i] = di * di * xw3[i] + bias[0];
    }
}

// lane = feature: 16 consecutive threads share one edge -> coalesced 64B
// gather of xw[src] and coalesced atomic burst on out[dst].
__global__ void k_scatter16(const int* __restrict__ src, const int* __restrict__ dst,
                            const float* __restrict__ dinv, const float* __restrict__ xw,
                            float* __restrict__ out, int n_edges) {
    long long t = (long long)blockIdx.x * blockDim.x + threadIdx.x;
    int e = (int)(t >> 4), f = (int)(t & 15);
    if (e >= n_edges) return;
    int s = src[e], d = dst[e];
    float nrm = dinv[s] * dinv[d];
    atomicAdd(&out[(size_t)d * 16 + f], nrm * xw[(size_t)s * 16 + f]);
}

__global__ void k_scatter1(const int* __restrict__ src, const int* __restrict__ dst,
                           const float* __restrict__ dinv, const float* __restrict__ xw3,
                           float* __restrict__ out, int n_edges) {
    int e = blockIdx.x * blockDim.x + threadIdx.x;
    if (e >= n_edges) return;
    int s = src[e], d = dst[e];
    atomicAdd(&out[d], dinv[s] * dinv[d] * xw3[s]);
}

// ---------------------------------------------------------------------------
extern "C" void kernel_launch(void* const* d_in, const int* in_sizes, int n_in,
                              void* d_out, int out_size, void* d_ws, size_t ws_size,
                              hipStream_t stream) {
    const float* x  = (const float*)d_in[0];
    const int*   ei = (const int*)d_in[1];   // [2, E] flat: row0=src, row1=dst
    const float* W1 = (const float*)d_in[2];
    const float* b1 = (const float*)d_in[3];
    const float* W2 = (const float*)d_in[4];
    const float* b2 = (const float*)d_in[5];
    const float* W3 = (const float*)d_in[6];
    const float* b3 = (const float*)d_in[7];
    float* out = (float*)d_out;

    const int N = in_sizes[0] / 2;           // 250000 nodes (x is [N,2])
    const int E = in_sizes[1] / 2;           // 4000000 edges
    const int* src = ei;
    const int* dst = ei + E;

    // Workspace layout (floats): dinv[N] | bufXW[N*16] | bufH[N*16]
    float* ws    = (float*)d_ws;
    float* dinv  = ws;
    float* bufXW = ws + N;
    float* bufH  = bufXW + (size_t)N * 16;
    float* xw3   = bufXW;                    // reuse: xw1/xw2 dead by layer 3

    const int B = 256;
    const int gN   = (N + B - 1) / B;
    const int gE   = (E + B - 1) / B;
    const int gN16 = (int)(((long long)N * 16 + B - 1) / B);
    const int gE16 = (int)(((long long)E * 16 + B - 1) / B);
    const int tiles = (N + 15) / 16;
    const int gT   = (tiles + WAVES_PER_BLOCK - 1) / WAVES_PER_BLOCK;

    // --- normalization ---
    k_deg_init <<<gN, B, 0, stream>>>(dinv, N);
    k_deg_accum<<<gE, B, 0, stream>>>(dst, dinv, E);
    k_dinv     <<<gN, B, 0, stream>>>(dinv, N);

    // --- layer 1: x[N,2] -> bufH[N,16] (ReLU deferred to layer-2 read) ---
    k_gemm_in2 <<<gT, B, 0, stream>>>(x, W1, bufXW, N, tiles);
    k_init16   <<<gN16, B, 0, stream>>>(bufXW, dinv, b1, bufH, N);
    k_scatter16<<<gE16, B, 0, stream>>>(src, dst, dinv, bufXW, bufH, E);

    // --- layer 2: relu(bufH) -> bufH[N,16] (ReLU deferred to layer-3 read) ---
    k_gemm16<true><<<gT, B, 0, stream>>>(bufH, W2, bufXW, N, tiles);
    k_init16   <<<gN16, B, 0, stream>>>(bufXW, dinv, b2, bufH, N);
    k_scatter16<<<gE16, B, 0, stream>>>(src, dst, dinv, bufXW, bufH, E);

    // --- layer 3: relu(bufH)[N,16] -> out[N,1] ---
    k_gemm_out1<<<gN, B, 0, stream>>>(bufH, W3, xw3, N);
    k_init1    <<<gN, B, 0, stream>>>(xw3, dinv, b3, out, N);
    k_scatter1 <<<gE, B, 0, stream>>>(src, dst, dinv, xw3, out, E);
}